// AnatomicalSSIMLoss_9552007266568
// MI455X (gfx1250) — compile-verified
//
#include <hip/hip_runtime.h>
#include <hip/hip_bf16.h>
#include <stdint.h>

// ---------------- problem constants (from reference) ----------------
#define BATCH 128
#define HH 384
#define WW 384
#define OH 378               // H - WIN + 1
#define OW 378
#define WPR 12               // 384 / 32 words per bit-packed row
#define WORDS_PER_IMG (HH * WPR)   // 4608 words = 18KB per image
#define THRESH 5e-05f

#define TILES_X 24           // ceil(378/16)
#define TILES_Y 24
#define TILES_PER_IMG (TILES_X * TILES_Y)
#define TOTAL_TILES (BATCH * TILES_PER_IMG)       // 73728
#define WAVES_PER_BLOCK 4
#define MAIN_BLOCKS (TOTAL_TILES / WAVES_PER_BLOCK) // 18432

typedef __attribute__((ext_vector_type(2))) float v2f;
typedef __attribute__((ext_vector_type(8))) float v8f;

// =====================================================================
// Kernel 1: mask = erode29(dilate31(erode3(Y > THRESH)))
// One block per image; mask kept as bit-packed rows entirely in LDS.
// Erosion padding = identity(1), dilation padding = identity(0), matching
// lax.reduce_window SAME padding with +/-inf init.
// =====================================================================
__device__ inline void hpass_bits(const uint32_t* src, uint32_t* dst,
                                  int R, bool erode, int tid) {
    const uint32_t ID = erode ? 0xFFFFFFFFu : 0u;
    for (int w = tid; w < WORDS_PER_IMG; w += 256) {
        int wc = w % WPR;
        uint32_t x  = src[w];
        uint32_t p  = (wc > 0)       ? src[w - 1] : ID;
        uint32_t nx = (wc < WPR - 1) ? src[w + 1] : ID;
        uint32_t acc = x;
        for (int s = 1; s <= R; ++s) {
            uint32_t right = (x >> s) | (nx << (32 - s)); // col + s
            uint32_t left  = (x << s) | (p  >> (32 - s)); // col - s
            if (erode) acc &= (right & left);
            else       acc |= (right | left);
        }
        dst[w] = acc;
    }
}

__device__ inline void vpass_bits(const uint32_t* src, uint32_t* dst,
                                  int R, bool erode, int tid) {
    for (int w = tid; w < WORDS_PER_IMG; w += 256) {
        int r = w / WPR;
        uint32_t acc = src[w];
        for (int d = 1; d <= R; ++d) {
            if (r - d >= 0) { uint32_t v = src[w - d * WPR]; acc = erode ? (acc & v) : (acc | v); }
            if (r + d < HH) { uint32_t v = src[w + d * WPR]; acc = erode ? (acc & v) : (acc | v); }
        }
        dst[w] = acc;
    }
}

__global__ __launch_bounds__(256)
void mask_kernel(const float* __restrict__ Y, uint32_t* __restrict__ maskbits) {
    __shared__ uint32_t bufA[WORDS_PER_IMG];
    __shared__ uint32_t bufB[WORDS_PER_IMG];
    const int b    = blockIdx.x;
    const int tid  = threadIdx.x;
    const int lane = tid & 31;
    const int wv   = tid >> 5;      // 8 waves
    const float* Yb = Y + (size_t)b * HH * WW;

    // threshold -> bits via wave32 ballot (coalesced 32-float row segments)
    for (int w = wv; w < WORDS_PER_IMG; w += 8) {
        int r = w / WPR, wc = w % WPR;
        float y = Yb[r * WW + wc * 32 + lane];
        unsigned long long bal = __ballot(y > THRESH);
        if (lane == 0) bufA[w] = (uint32_t)bal;
    }
    __syncthreads();

    hpass_bits(bufA, bufB, 1,  true,  tid);  __syncthreads();  // erode 3 (H)
    vpass_bits(bufB, bufA, 1,  true,  tid);  __syncthreads();  // erode 3 (V)
    hpass_bits(bufA, bufB, 15, false, tid);  __syncthreads();  // dilate 31 (H)
    vpass_bits(bufB, bufA, 15, false, tid);  __syncthreads();  // dilate 31 (V)
    hpass_bits(bufA, bufB, 14, true,  tid);  __syncthreads();  // erode 29 (H)
    vpass_bits(bufB, bufA, 14, true,  tid);  __syncthreads();  // erode 29 (V)

    for (int w = tid; w < WORDS_PER_IMG; w += 256)
        maskbits[(size_t)b * WORDS_PER_IMG + w] = bufA[w];
}

// =====================================================================
// Kernel 2: fused masked 7x7 box sums + SSIM map + partial reduction.
// One wave per 16x16 output tile; vertical 7-tap sums via
// V_WMMA_F32_16X16X4_F32:  O(16x16) = Wv(16x24 band of ones) * H(24x16),
// where H rows 22..23 are zero-padded so all B-operand LDS loads are
// unconditional (no EXEC divergence in the WMMA stream).
// Each wave uses a private LDS slice -> no block barriers until the
// final partial-sum combine.
// =====================================================================
#define IT   22    // input tile (16 + WIN - 1)
#define ITP  23    // padded LDS row stride (odd -> bank-conflict free)
#define HROW 24    // Hs rows incl. 2 zero rows for K padding

__global__ __launch_bounds__(128)
void ssim_kernel(const float* __restrict__ X, const float* __restrict__ Y,
                 const float* __restrict__ data_range,
                 const uint32_t* __restrict__ maskbits,
                 float* __restrict__ partials) {
    __shared__ float xs[WAVES_PER_BLOCK][IT * ITP];
    __shared__ float ys[WAVES_PER_BLOCK][IT * ITP];
    __shared__ float Hs[WAVES_PER_BLOCK][5][HROW * 16];
    __shared__ float wsum[WAVES_PER_BLOCK];

    const int tid  = threadIdx.x;
    const int lane = tid & 31;
    const int wv   = tid >> 5;
    const int tile = blockIdx.x * WAVES_PER_BLOCK + wv;   // < TOTAL_TILES
    const int b    = tile / TILES_PER_IMG;
    const int rem  = tile % TILES_PER_IMG;
    const int r0   = (rem / TILES_X) * 16;
    const int c0   = (rem % TILES_X) * 16;

    const float*    Xb = X + (size_t)b * HH * WW;
    const float*    Yb = Y + (size_t)b * HH * WW;
    const uint32_t* Mb = maskbits + (size_t)b * WORDS_PER_IMG;

    float* xsw = xs[wv];
    float* ysw = ys[wv];
    float (*Hw)[HROW * 16] = Hs[wv];

    // ---- zero the two K-padding rows (22,23) of every map ----
    #pragma unroll
    for (int map = 0; map < 5; ++map)
        Hw[map][IT * 16 + lane] = 0.f;          // elements 352..383

    // ---- stage masked 22x22 input tile into LDS (zero-pad OOB) ----
    for (int e = lane; e < IT * IT; e += 32) {
        int rr = e / IT, cc = e % IT;
        int ra = r0 + rr, ca = c0 + cc;
        float xvv = 0.f, yvv = 0.f;
        if (ra < HH && ca < WW) {
            uint32_t mw = Mb[ra * WPR + (ca >> 5)];
            float m = (float)((mw >> (ca & 31)) & 1u);
            xvv = Xb[ra * WW + ca] * m;
            yvv = Yb[ra * WW + ca] * m;
        }
        xsw[rr * ITP + cc] = xvv;
        ysw[rr * ITP + cc] = yvv;
    }

    // ---- horizontal 7-tap sums: Hs[map][k][n], k=0..21, n=0..15 ----
    for (int e = lane; e < IT * 16; e += 32) {   // 352 = 11 * 32 exactly
        int k = e >> 4, n = e & 15;
        const float* xr = &xsw[k * ITP + n];
        const float* yr = &ysw[k * ITP + n];
        float sx = 0.f, sy = 0.f, sxx = 0.f, syy = 0.f, sxy = 0.f;
        #pragma unroll
        for (int t = 0; t < 7; ++t) {
            float xvv = xr[t], yvv = yr[t];
            sx += xvv; sy += yvv;
            sxx = fmaf(xvv, xvv, sxx);
            syy = fmaf(yvv, yvv, syy);
            sxy = fmaf(xvv, yvv, sxy);
        }
        Hw[0][e] = sx; Hw[1][e] = sy; Hw[2][e] = sxx; Hw[3][e] = syy; Hw[4][e] = sxy;
    }

    // ---- vertical 7-tap via WMMA f32 16x16x4 ----
    const int koff = (lane < 16) ? 0 : 2;   // A/B K-halves across lane halves
    const int m    = lane & 15;             // A row
    const int n    = lane & 15;             // B/C column

    // A = banded ones Wv[m][k] = (m <= k <= m+6); with m <= 15 this is
    // automatically 0 for k >= 22, and B rows 22..23 are zero anyway.
    v2f Avec[6];
    #pragma unroll
    for (int j = 0; j < 6; ++j) {
        int k0 = 4 * j + koff;
        Avec[j].x = (k0     >= m && k0     <= m + 6) ? 1.f : 0.f;
        Avec[j].y = (k0 + 1 >= m && k0 + 1 <= m + 6) ? 1.f : 0.f;
    }

    v8f Cacc[5];
    #pragma unroll
    for (int map = 0; map < 5; ++map) {
        v8f c = {};
#if __has_builtin(__builtin_amdgcn_wmma_f32_16x16x4_f32)
        #pragma unroll
        for (int j = 0; j < 6; ++j) {
            int k0 = 4 * j + koff;
            v2f bvec;                                   // unconditional loads
            bvec.x = Hw[map][k0 * 16 + n];              // (ds_load_2addr pair)
            bvec.y = Hw[map][(k0 + 1) * 16 + n];
            // D = A * B + C  (f32, exact precision path)
            c = __builtin_amdgcn_wmma_f32_16x16x4_f32(false, Avec[j], false, bvec,
                                                      (short)0, c, false, false);
        }
#else
        // scalar fallback (correctness path if builtin missing)
        #pragma unroll
        for (int k = 0; k < 8; ++k) {
            int ml = k + ((lane < 16) ? 0 : 8);
            float s = 0.f;
            #pragma unroll
            for (int t = 0; t < 7; ++t) s += Hw[map][(ml + t) * 16 + n];
            c[k] = s;
        }
#endif
        Cacc[map] = c;
    }

    // ---- elementwise SSIM in the WMMA C layout + accumulate ----
    const float dr  = data_range[b];
    const float C1  = (0.01f * dr) * (0.01f * dr);
    const float C2  = (0.03f * dr) * (0.03f * dr);
    const float inv_np = 1.0f / 49.0f;
    const float cov    = 49.0f / 48.0f;

    float acc = 0.f;
    #pragma unroll
    for (int k = 0; k < 8; ++k) {
        int ml = k + ((lane < 16) ? 0 : 8);     // C layout: vgpr k -> M=k / k+8
        int om = r0 + ml, on = c0 + n;
        if (om < OH && on < OW) {
            float ux  = Cacc[0][k] * inv_np;
            float uy  = Cacc[1][k] * inv_np;
            float uxx = Cacc[2][k] * inv_np;
            float uyy = Cacc[3][k] * inv_np;
            float uxy = Cacc[4][k] * inv_np;
            float vx  = cov * (uxx - ux * ux);
            float vy  = cov * (uyy - uy * uy);
            float vxy = cov * (uxy - ux * uy);
            float A1  = 2.f * ux * uy + C1;
            float A2  = 2.f * vxy + C2;
            float B1  = ux * ux + uy * uy + C1;
            float B2  = vx + vy + C2;
            acc += (A1 * A2) / (B1 * B2);
        }
    }

    // wave32 shuffle reduction, then deterministic per-block combine
    #pragma unroll
    for (int off = 16; off > 0; off >>= 1) acc += __shfl_xor(acc, off, 32);
    if (lane == 0) wsum[wv] = acc;
    __syncthreads();
    if (tid == 0) {
        float s = 0.f;
        for (int w = 0; w < WAVES_PER_BLOCK; ++w) s += wsum[w];
        partials[blockIdx.x] = s;
    }
}

// =====================================================================
// Kernel 3: deterministic final reduction -> 1 - mean(S)
// =====================================================================
__global__ __launch_bounds__(256)
void finalize_kernel(const float* __restrict__ partials, int nparts,
                     float* __restrict__ out) {
    __shared__ double sd[256];
    double s = 0.0;
    for (int i = threadIdx.x; i < nparts; i += 256) s += (double)partials[i];
    sd[threadIdx.x] = s;
    __syncthreads();
    for (int off = 128; off > 0; off >>= 1) {
        if ((int)threadIdx.x < off) sd[threadIdx.x] += sd[threadIdx.x + off];
        __syncthreads();
    }
    if (threadIdx.x == 0) {
        const double N = (double)BATCH * (double)OH * (double)OW;
        out[0] = (float)(1.0 - sd[0] / N);
    }
}

// =====================================================================
extern "C" void kernel_launch(void* const* d_in, const int* in_sizes, int n_in,
                              void* d_out, int out_size, void* d_ws, size_t ws_size,
                              hipStream_t stream) {
    const float* X  = (const float*)d_in[0];
    const float* Y  = (const float*)d_in[1];
    const float* dr = (const float*)d_in[2];

    // workspace layout: [mask bits: 128*4608 u32 = 2.25MB][partials: 18432 f32]
    uint32_t* maskbits = (uint32_t*)d_ws;
    float*    partials = (float*)((char*)d_ws + (size_t)BATCH * WORDS_PER_IMG * sizeof(uint32_t));
    float*    out      = (float*)d_out;

    mask_kernel<<<BATCH, 256, 0, stream>>>(Y, maskbits);
    ssim_kernel<<<MAIN_BLOCKS, 128, 0, stream>>>(X, Y, dr, maskbits, partials);
    finalize_kernel<<<1, 256, 0, stream>>>(partials, MAIN_BLOCKS, out);
}